// _SecondOrderAttention_1305670058081
// MI455X (gfx1250) — compile-verified
//
#include <hip/hip_runtime.h>
#include <hip/hip_bf16.h>
#include <math.h>

// Problem constants (from reference setup_inputs)
#define BB 16
#define CC 256
#define RR 16
#define HN 16384          // H*W = 128*128
#define HN4 (HN / 4)      // float4 elements per channel row

typedef float v2f __attribute__((ext_vector_type(2)));
typedef float v8f __attribute__((ext_vector_type(8)));

__device__ __forceinline__ float wave_reduce(float v) {
  #pragma unroll
  for (int off = 16; off; off >>= 1) v += __shfl_xor(v, off, 32);
  return v;
}

// ---------------------------------------------------------------------------
// Pass 0: zero the rowsum accumulator (it is built with atomics)
// ---------------------------------------------------------------------------
__global__ void k_zero(float* __restrict__ p, int n) {
  int i = blockIdx.x * blockDim.x + threadIdx.x;
  if (i < n) p[i] = 0.0f;
}

// ---------------------------------------------------------------------------
// Pass 1: single read of x produces
//   rowsum[b,c] = sum_n x[b,c,n]        (global f32 atomics, 65K total)
//   colsum[b,n] = sum_c x[b,c,n]        (direct store, block owns pixel range)
// Grid: BB * 16 blocks, 256 threads; each block covers 1024 pixels x 256 ch.
// ---------------------------------------------------------------------------
__global__ void k_pass1(const float* __restrict__ x,
                        float* __restrict__ rowsum,
                        float* __restrict__ colsum) {
  __shared__ float ldsRow[CC][8];            // per-channel, per-wave partials (8 KB)
  const int tid  = threadIdx.x;
  const int wave = tid >> 5, lane = tid & 31;
  const int b    = blockIdx.x >> 4;          // 16 pixel-blocks per batch
  const int nb   = blockIdx.x & 15;
  const int n4   = nb * 256 + tid;           // float4 pixel index

  const float4* xb = (const float4*)(x + (size_t)b * CC * HN) + n4;
  float4 colacc = make_float4(0.f, 0.f, 0.f, 0.f);

  #pragma unroll 2
  for (int c = 0; c < CC; ++c) {
    float4 v = xb[(size_t)c * HN4];
    colacc.x += v.x; colacc.y += v.y; colacc.z += v.z; colacc.w += v.w;
    float rp = wave_reduce((v.x + v.y) + (v.z + v.w));
    if (lane == 0) ldsRow[c][wave] = rp;
  }

  ((float4*)(colsum + (size_t)b * HN))[n4] = colacc;
  __syncthreads();

  float r = 0.f;
  #pragma unroll
  for (int w = 0; w < 8; ++w) r += ldsRow[tid][w];
  atomicAdd(&rowsum[b * CC + tid], r);       // tid == channel index
}

// ---------------------------------------------------------------------------
// Pass 2: v[b,c] = (sum_n x[b,c,n]*s[b,n] - mu[b,c]*sum_n s[b,n]) / ((N-1)*C)
//         with s[b,n] = colsum[b,n] - musum[b],  musum[b] = sum_c rowsum / N
// Grid: BB*CC blocks (one per channel row), 256 threads.
// colsum row (64KB/batch) stays hot in L2 across the 256 blocks of a batch.
// ---------------------------------------------------------------------------
__global__ void k_pass2(const float* __restrict__ x,
                        const float* __restrict__ rowsum,
                        const float* __restrict__ colsum,
                        float* __restrict__ vout) {
  __shared__ float red[16];
  const int tid  = threadIdx.x;
  const int wave = tid >> 5, lane = tid & 31;
  const int b = blockIdx.x >> 8;
  const int c = blockIdx.x & 255;

  // musum[b]: block-reduce the 256 rowsum entries of this batch
  float t = wave_reduce(rowsum[b * CC + tid]);
  if (lane == 0) red[wave] = t;
  __syncthreads();
  float tot = 0.f;
  #pragma unroll
  for (int w = 0; w < 8; ++w) tot += red[w];
  const float musum = tot * (1.0f / (float)HN);
  __syncthreads();

  const float4* x4 = (const float4*)(x + (size_t)(b * CC + c) * HN);
  const float4* s4 = (const float4*)(colsum + (size_t)b * HN);
  float dot = 0.f, ssum = 0.f;
  #pragma unroll 4
  for (int i = tid; i < HN4; i += 256) {
    float4 xv = x4[i], cv = s4[i];
    float sx = cv.x - musum, sy = cv.y - musum, sz = cv.z - musum, sw = cv.w - musum;
    dot = fmaf(xv.x, sx, dot);
    dot = fmaf(xv.y, sy, dot);
    dot = fmaf(xv.z, sz, dot);
    dot = fmaf(xv.w, sw, dot);
    ssum += (sx + sy) + (sz + sw);
  }
  dot = wave_reduce(dot);
  ssum = wave_reduce(ssum);
  if (lane == 0) { red[wave] = dot; red[8 + wave] = ssum; }
  __syncthreads();
  if (tid == 0) {
    float D = 0.f, S = 0.f;
    #pragma unroll
    for (int w = 0; w < 8; ++w) { D += red[w]; S += red[8 + w]; }
    float mu = rowsum[b * CC + c] * (1.0f / (float)HN);
    vout[b * CC + c] = (D - mu * S) * (1.0f / ((float)(HN - 1) * (float)CC));
  }
}

// ---------------------------------------------------------------------------
// Pass 3: the SE-MLP on V_WMMA_F32_16X16X4_F32 (f32 in, f32 out — exact).
//   h1   = relu(v @ w1^T + b1)     : M=16, N=16, K=256  -> 64 chained WMMAs
//   gate = sigmoid(h1 @ w2^T + b2) : M=16, N=256, K=16  -> 16 tiles x 4 WMMAs
// One wave32 block; EXEC all ones as WMMA requires.
// A-frag (16x4 f32): lanes 0-15 hold M=lane, K=k0{0,1}; lanes 16-31 K=k0{2,3}.
// B-frag (4x16 f32): lanes 0-15 hold N=lane, K=k0{0,1}; lanes 16-31 K=k0{2,3}.
// D (16x16 f32): VGPR r -> M = r + 8*(lane>=16), N = lane&15.
// ---------------------------------------------------------------------------
__global__ void k_mlp_wmma(const float* __restrict__ v,
                           const float* __restrict__ w1,
                           const float* __restrict__ b1,
                           const float* __restrict__ w2,
                           const float* __restrict__ b2,
                           float* __restrict__ gate) {
  __shared__ float h1s[16 * 16];
  const int lane  = threadIdx.x;
  const int idx   = lane & 15;        // M row for A-frags, N col for B-frags
  const int hi    = lane >> 4;        // which K half this half-wave supplies
  const int khalf = hi << 1;          // 0 or 2

  // ---- GEMM 1: h1pre = v @ w1^T, K = 256 ----
  v8f acc = {};
  for (int kk = 0; kk < CC; kk += 4) {
    const int k0 = kk + khalf;
    v2f a, bf;
    a[0]  = v[idx * CC + k0];
    a[1]  = v[idx * CC + k0 + 1];
    bf[0] = w1[idx * CC + k0];        // B[k][n] = w1[n][k]
    bf[1] = w1[idx * CC + k0 + 1];
    acc = __builtin_amdgcn_wmma_f32_16x16x4_f32(false, a, false, bf,
                                                (short)0, acc, false, false);
  }
  // bias + relu, re-stripe D-layout -> row-major LDS for the next A operand
  #pragma unroll
  for (int r = 0; r < 8; ++r) {
    int m = r + 8 * hi;               // batch row
    float val = acc[r] + b1[idx];     // idx == hidden unit (N)
    h1s[m * 16 + idx] = val > 0.f ? val : 0.f;
  }
  __syncthreads();

  // ---- GEMM 2: gate = sigmoid(h1 @ w2^T + b2), K = 16, 16 N-tiles ----
  for (int tile = 0; tile < 16; ++tile) {
    v8f acc2 = {};
    #pragma unroll
    for (int kk = 0; kk < 16; kk += 4) {
      const int k0 = kk + khalf;
      v2f a, bf;
      a[0] = h1s[idx * 16 + k0];
      a[1] = h1s[idx * 16 + k0 + 1];
      const int ccol = tile * 16 + idx;
      bf[0] = w2[ccol * RR + k0];     // B[k][n] = w2[n][k]
      bf[1] = w2[ccol * RR + k0 + 1];
      acc2 = __builtin_amdgcn_wmma_f32_16x16x4_f32(false, a, false, bf,
                                                   (short)0, acc2, false, false);
    }
    #pragma unroll
    for (int r = 0; r < 8; ++r) {
      int m = r + 8 * hi;
      int col = tile * 16 + idx;
      float z = acc2[r] + b2[col];
      gate[m * CC + col] = 1.0f / (1.0f + __expf(-z));
    }
  }
}

// ---------------------------------------------------------------------------
// Pass 4: out = x * gate[b,c]   (read 256MiB + write 256MiB, float4)
// gate index = float4_index / 4096 — uniform per block (blocks never straddle
// a channel), so the compiler scalarizes the gate load.
// ---------------------------------------------------------------------------
__global__ void k_scale(const float* __restrict__ x,
                        const float* __restrict__ gate,
                        float* __restrict__ out) {
  const size_t i = (size_t)blockIdx.x * blockDim.x + threadIdx.x;  // float4 idx
  const float g = gate[i >> 12];                                   // HN/4 = 4096
  const float4 xv = ((const float4*)x)[i];
  ((float4*)out)[i] = make_float4(xv.x * g, xv.y * g, xv.z * g, xv.w * g);
}

// ---------------------------------------------------------------------------
extern "C" void kernel_launch(void* const* d_in, const int* in_sizes, int n_in,
                              void* d_out, int out_size, void* d_ws, size_t ws_size,
                              hipStream_t stream) {
  const float* x  = (const float*)d_in[0];
  const float* w1 = (const float*)d_in[1];
  const float* b1 = (const float*)d_in[2];
  const float* w2 = (const float*)d_in[3];
  const float* b2 = (const float*)d_in[4];
  float* out = (float*)d_out;

  // workspace layout (floats): rowsum[4096] | colsum[262144] | v[4096] | gate[4096]
  float* ws     = (float*)d_ws;
  float* rowsum = ws;
  float* colsum = ws + BB * CC;
  float* vdesc  = colsum + BB * HN;
  float* gate   = vdesc + BB * CC;

  k_zero<<<(BB * CC + 255) / 256, 256, 0, stream>>>(rowsum, BB * CC);
  k_pass1<<<BB * 16, 256, 0, stream>>>(x, rowsum, colsum);
  k_pass2<<<BB * CC, 256, 0, stream>>>(x, rowsum, colsum, vdesc);
  k_mlp_wmma<<<1, 32, 0, stream>>>(vdesc, w1, b1, w2, b2, gate);
  k_scale<<<(BB * CC * HN / 4) / 256, 256, 0, stream>>>(x, gate, out);
}